// ExpectedCalibrationError_13769665151538
// MI455X (gfx1250) — compile-verified
//
#include <hip/hip_runtime.h>
#include <hip/hip_bf16.h>
#include <stdint.h>

#define ECE_NBINS 15

// Fixed-point scale for per-wave confidence accumulation (bits [31:0] of pack).
#define CONF_SCALE 1048576.0f            // 2^20
#define CONF_INV_SCALE (1.0 / 1048576.0)

// Packed per-element increment layout (u64, single ds_add_u64):
//   [31:0]  : round(c * 2^20)   (per-wave sum <= ~2^31, no carry into [32])
//   [47:32] : +1 count          (per-wave sum <= ~2051)
//   [63:48] : +1 if p==t        (per-wave sum <= count)
// Field widths hold as long as each wave processes <= 2048 (+3 tail) elements;
// the launcher sizes the grid to guarantee that.

// Workspace layout (bytes). Zeroed by ece_init_kernel at every launch.
//   [0,120)   : u64 pack[15]  -- global counts in [31:0], correct in [63:32]
//   [128,248) : f64 conf[15]  -- per-bin confidence sums (already descaled)
//   [256,264) : u64 total     -- sum of mask
#define WS_PACK_OFF   0
#define WS_CONF_OFF   128
#define WS_TOTAL_OFF  256
#define WS_USED_BYTES 264

__global__ void ece_init_kernel(uint32_t* __restrict__ ws) {
  uint32_t i = threadIdx.x;
  if (i < WS_USED_BYTES / 4) ws[i] = 0u;
}

__device__ __forceinline__ void ece_accum(float c, int p, int t, int m,
                                          unsigned long long* __restrict__ pack_row,
                                          uint32_t& total_m) {
  total_m += (uint32_t)m;                           // total = sum(mask), not in_range-gated
  int idx = (int)ceilf(c * (float)ECE_NBINS) - 1;   // same f32 math as reference
  idx = idx < 0 ? 0 : (idx > ECE_NBINS - 1 ? ECE_NBINS - 1 : idx);  // -> v_med3
  bool in_range = (c > 0.0f) && (c <= 1.0f);
  if ((m != 0) && in_range) {
    uint32_t cfix = (uint32_t)__builtin_fmaf(c, CONF_SCALE, 0.5f);  // round-to-nearest-ish
    uint32_t hi   = 1u | ((p == t) ? 0x10000u : 0u);  // count bit + correct bit
    unsigned long long pv = (unsigned long long)cfix |
                            ((unsigned long long)hi << 32);
    atomicAdd(&pack_row[idx], pv);   // single ds_add_u64, wave-private row
  }
}

__global__ __launch_bounds__(256) void ece_hist_kernel(
    const float* __restrict__ conf, const int* __restrict__ pred,
    const int* __restrict__ targ, const int* __restrict__ mask,
    uint8_t* __restrict__ ws, long long n) {
  // 256 threads = 8 wave32s; each wave owns a private 16-slot histogram row,
  // so ds_add_u64 only serializes on same-bin lanes within one wave.
  __shared__ unsigned long long s_pack[8][16];
  __shared__ uint32_t s_total;

  const int tid  = threadIdx.x;
  const int wave = tid >> 5;

  for (int i = tid; i < 8 * 16; i += 256)
    (&s_pack[0][0])[i] = 0ull;
  if (tid == 0) s_total = 0u;
  __syncthreads();

  unsigned long long* pack_row = s_pack[wave];
  uint32_t total_m = 0;

  const uint32_t n4 = (uint32_t)(n >> 2);           // vec4 count; arrays are 128 MB
  const float4* __restrict__ c4 = reinterpret_cast<const float4*>(conf);
  const int4*   __restrict__ p4 = reinterpret_cast<const int4*>(pred);
  const int4*   __restrict__ t4 = reinterpret_cast<const int4*>(targ);
  const int4*   __restrict__ m4 = reinterpret_cast<const int4*>(mask);

  const uint32_t stride = gridDim.x * 256u;         // 32-bit index math -> scale_offset
  for (uint32_t i = blockIdx.x * 256u + (uint32_t)tid; i < n4; i += stride) {
    // Speculative prefetch of the next grid-stride tile (global_prefetch_b8;
    // OOB translations silently dropped). Window = 4 streams * 16*stride bytes,
    // comfortably inside the 192 MB L2.
    __builtin_prefetch(&c4[i + stride], 0, 0);
    __builtin_prefetch(&p4[i + stride], 0, 0);
    __builtin_prefetch(&t4[i + stride], 0, 0);
    __builtin_prefetch(&m4[i + stride], 0, 0);

    float4 c = c4[i];   // global_load_b128 x4: full-width streaming
    int4   p = p4[i];
    int4   t = t4[i];
    int4   m = m4[i];
    ece_accum(c.x, p.x, t.x, m.x, pack_row, total_m);
    ece_accum(c.y, p.y, t.y, m.y, pack_row, total_m);
    ece_accum(c.z, p.z, t.z, m.z, pack_row, total_m);
    ece_accum(c.w, p.w, t.w, m.w, pack_row, total_m);
  }

  // Scalar tail (n % 4) — none for 8192*4096, kept for generality.
  if (blockIdx.x == 0 && tid == 0) {
    for (long long i = (long long)n4 << 2; i < n; ++i)
      ece_accum(conf[i], pred[i], targ[i], mask[i], pack_row, total_m);
  }

  atomicAdd(&s_total, total_m);
  __syncthreads();

  // Unpack fields per wave (8-wave sums would overflow the packed fields),
  // then one packed u64 + one f64 global atomic per bin per block.
  if (tid < ECE_NBINS) {
    uint32_t cnt = 0, cor = 0;
    double csum = 0.0;
    #pragma unroll
    for (int w = 0; w < 8; ++w) {
      unsigned long long v = s_pack[w][tid];
      csum += (double)(uint32_t)(v & 0xffffffffull);
      cnt  += (uint32_t)((v >> 32) & 0xffffull);
      cor  += (uint32_t)(v >> 48);
    }
    atomicAdd(reinterpret_cast<unsigned long long*>(ws + WS_PACK_OFF) + tid,
              (unsigned long long)cnt | ((unsigned long long)cor << 32));
    atomicAdd(reinterpret_cast<double*>(ws + WS_CONF_OFF) + tid,
              csum * CONF_INV_SCALE);
  }
  if (tid == 0)
    atomicAdd(reinterpret_cast<unsigned long long*>(ws + WS_TOTAL_OFF),
              (unsigned long long)s_total);
}

__global__ void ece_final_kernel(const uint8_t* __restrict__ ws,
                                 float* __restrict__ out) {
  const unsigned long long* pack =
      reinterpret_cast<const unsigned long long*>(ws + WS_PACK_OFF);
  const double* confs = reinterpret_cast<const double*>(ws + WS_CONF_OFF);
  const float total =
      (float)*reinterpret_cast<const unsigned long long*>(ws + WS_TOTAL_OFF);
  float ece = 0.0f;
  for (int b = 0; b < ECE_NBINS; ++b) {
    unsigned long long pv = pack[b];
    float cnt = (float)(uint32_t)(pv & 0xffffffffull);
    float cor = (float)(uint32_t)(pv >> 32);
    float cs  = (float)confs[b];
    if (cnt > 0.0f) {
      float safe = fmaxf(cnt, 1.0f);
      ece += fabsf(cs / safe - cor / safe) * cnt / total;
    }
  }
  out[0] = ece;
}

extern "C" void kernel_launch(void* const* d_in, const int* in_sizes, int n_in,
                              void* d_out, int out_size, void* d_ws, size_t ws_size,
                              hipStream_t stream) {
  (void)n_in; (void)out_size; (void)ws_size;
  const float* conf = (const float*)d_in[0];
  const int*   pred = (const int*)d_in[1];
  const int*   targ = (const int*)d_in[2];
  const int*   mask = (const int*)d_in[3];
  const long long n = (long long)in_sizes[0];

  ece_init_kernel<<<1, 128, 0, stream>>>((uint32_t*)d_ws);

  // Size the grid so each wave handles <= 2048 elements (512 vec4 groups):
  // guarantees the packed per-wave fixed-point fields cannot overflow, and
  // for n=33.5M gives 4096 blocks (~1M threads, 8 vec4 iters each) — enough
  // waves in flight to saturate 23.3 TB/s.
  long long n4 = n >> 2;
  long long blocks_ll = (n4 + 2047) / 2048;   // 2048 vec4 per block = 1024 elem/wave
  if (blocks_ll < 1) blocks_ll = 1;
  int blocks = (int)blocks_ll;

  ece_hist_kernel<<<blocks, 256, 0, stream>>>(conf, pred, targ, mask,
                                              (uint8_t*)d_ws, n);
  ece_final_kernel<<<1, 1, 0, stream>>>((const uint8_t*)d_ws, (float*)d_out);
}